// LSTM_80522046865747
// MI455X (gfx1250) — compile-verified
//
#include <hip/hip_runtime.h>
#include <hip/hip_bf16.h>
#include <math.h>
#include <stdint.h>

// Problem dims
#define B_DIM   256
#define C_DIM   4096
#define VOCAB   32000
#define KSTEP   32          // K-chunk staged in LDS per buffer
#define NCHUNK  (C_DIM / KSTEP)
#define A_STR   36          // padded LDS row stride (floats): 144B = 16B aligned, bank-spread
#define B_STR   36

typedef __attribute__((ext_vector_type(2))) float v2f;
typedef __attribute__((ext_vector_type(8))) float v8f;

// ---------------------------------------------------------------------------
// gfx1250 async global->LDS copy (ASYNCcnt-tracked), with safe fallback.
// Builtin prototype (from clang diagnostic): pointee is a 16-byte int vector,
// global src in AS1 ("__device__"), LDS dst in AS3.
// ---------------------------------------------------------------------------
#if defined(__gfx1250__) && __has_builtin(__builtin_amdgcn_global_load_async_to_lds_b128)
#define HAVE_ASYNC_LDS 1
#else
#define HAVE_ASYNC_LDS 0
#endif

#if HAVE_ASYNC_LDS
typedef int v16b __attribute__((vector_size(16)));
typedef v16b __attribute__((address_space(1)))* as1_b128_ptr;
typedef v16b __attribute__((address_space(3)))* as3_b128_ptr;
#endif

__device__ __forceinline__ void copy16_async(void* lds_dst, const void* gsrc) {
#if HAVE_ASYNC_LDS
    // AS1 global address == flat address numerically; AS3 offset == low 32 bits
    // of the flat LDS address (hardware aperture truncation).
    __builtin_amdgcn_global_load_async_to_lds_b128(
        (as1_b128_ptr)(unsigned long long)(uintptr_t)gsrc,
        (as3_b128_ptr)(unsigned int)(uintptr_t)lds_dst,
        0, 0);
#else
    *(float4*)lds_dst = *(const float4*)gsrc;
#endif
}

__device__ __forceinline__ void wait_async_all() {
#if HAVE_ASYNC_LDS
#if __has_builtin(__builtin_amdgcn_s_wait_asynccnt)
    __builtin_amdgcn_s_wait_asynccnt(0);
#else
    asm volatile("s_wait_asynccnt 0x0" ::: "memory");
#endif
#endif
}

__device__ __forceinline__ v8f wmma_f32(v2f a, v2f b, v8f c) {
    // D = A(16x4) * B(4x16) + C(16x16), fp32 matrix pipe
    return __builtin_amdgcn_wmma_f32_16x16x4_f32(
        false, a, false, b, (short)0, c, false, false);
}

__device__ __forceinline__ float sigmoidf_(float x) {
    return 1.0f / (1.0f + __expf(-x));
}

// ---------------------------------------------------------------------------
// Kernel 1: xc = concat([category, x, hidden], axis=1)  -> ws
// ---------------------------------------------------------------------------
__global__ __launch_bounds__(256) void concat_inputs(
    const float* __restrict__ cat, const float* __restrict__ x,
    const float* __restrict__ hid, float* __restrict__ xc)
{
    const int idx = blockIdx.x * 256 + threadIdx.x;  // float4 index, B*C/4 total
    const int b  = idx >> 10;                        // 1024 float4 per row
    const int j4 = idx & 1023;
    const int j  = j4 * 4;
    float4 v;
    if (j < 512) {
        v = ((const float4*)(cat + (size_t)b * 512))[j4];
    } else if (j < 2048) {
        v = ((const float4*)(x + (size_t)b * 1536))[(j - 512) >> 2];
    } else {
        v = ((const float4*)(hid + (size_t)b * 2048))[(j - 2048) >> 2];
    }
    ((float4*)(xc + (size_t)b * C_DIM))[j4] = v;
}

// ---------------------------------------------------------------------------
// Kernel 2: fused 4-gate GEMM + LSTM cell update.
// Workgroup: one 16-column slice, SAME columns of all 4 gates, full M=256
// (8 waves x 2 M-tiles). Weights stream from HBM exactly once.
// Double-buffered LDS, async global->LDS fills overlap WMMA compute.
// ---------------------------------------------------------------------------
__global__ __launch_bounds__(256) void lstm_gate_gemm(
    const float* __restrict__ xc,
    const float* __restrict__ Wf, const float* __restrict__ bf,
    const float* __restrict__ Wi, const float* __restrict__ bi,
    const float* __restrict__ Wc, const float* __restrict__ bc,
    const float* __restrict__ Wo, const float* __restrict__ bo,
    const float* __restrict__ cell,
    float* __restrict__ cell_new, float* __restrict__ hidden_new)
{
    __shared__ float A_lds[2][256 * A_STR];   // 256 batch rows x KSTEP, x2 buffers
    __shared__ float B_lds[2][64 * B_STR];    // 4 gates x 16 cols x KSTEP, x2

    const int tid    = threadIdx.x;
    const int wave   = tid >> 5;
    const int lane   = tid & 31;
    const int half   = lane >> 4;
    const int lane16 = lane & 15;
    const int n0     = blockIdx.x * 16;

    v8f acc[2][4];
    const v8f vzero = {0.f, 0.f, 0.f, 0.f, 0.f, 0.f, 0.f, 0.f};
#pragma unroll
    for (int t = 0; t < 2; ++t)
#pragma unroll
        for (int g = 0; g < 4; ++g) acc[t][g] = vzero;

    // B staging assignment: 64 weight rows (gate*16 + col), 4 threads/row
    const int rB = tid >> 2;      // 0..63
    const int qB = tid & 3;       // 8 floats (2x b128) each
    const float* Wsel = (rB < 16) ? Wf : (rB < 32) ? Wi : (rB < 48) ? Wc : Wo;
    const float* wrow = Wsel + (size_t)(n0 + (rB & 15)) * C_DIM;
    const float* aSrc = xc + (size_t)tid * C_DIM;     // A: thread t owns batch row t

    auto fill = [&](int buf, int k0) {
        const float* a = aSrc + k0;
        float* ad = &A_lds[buf][tid * A_STR];
#pragma unroll
        for (int i = 0; i < KSTEP / 4; ++i) copy16_async(ad + 4 * i, a + 4 * i);
        const float* b = wrow + k0 + qB * 8;
        float* bd = &B_lds[buf][rB * B_STR + qB * 8];
        copy16_async(bd, b);
        copy16_async(bd + 4, b + 4);
    };

    fill(0, 0);
    for (int i = 0; i < NCHUNK; ++i) {
        const int buf = i & 1;
        wait_async_all();                 // my fills of chunk i complete
        __syncthreads();                  // all fills of chunk i visible; all reads
                                          // of the other buffer (chunk i-1) done
        if (i + 1 < NCHUNK) fill(1 - buf, (i + 1) * KSTEP);

        const float* Ab = &A_lds[buf][0];
        const float* Bb = &B_lds[buf][0];
#pragma unroll
        for (int kk = 0; kk < KSTEP / 4; ++kk) {
            const int ka = kk * 4 + 2 * half;
            v2f a0, a1, bf4[4];
            const int ar0 = (wave * 32 + lane16) * A_STR + ka;
            const int ar1 = (wave * 32 + 16 + lane16) * A_STR + ka;
            a0.x = Ab[ar0];     a0.y = Ab[ar0 + 1];
            a1.x = Ab[ar1];     a1.y = Ab[ar1 + 1];
#pragma unroll
            for (int g = 0; g < 4; ++g) {
                const int br = (g * 16 + lane16) * B_STR + ka;
                bf4[g].x = Bb[br];
                bf4[g].y = Bb[br + 1];
            }
#pragma unroll
            for (int g = 0; g < 4; ++g) {
                acc[0][g] = wmma_f32(a0, bf4[g], acc[0][g]);
                acc[1][g] = wmma_f32(a1, bf4[g], acc[1][g]);
            }
        }
    }

    // epilogue: bias + gates + cell/hidden update, straight to d_out
    const int col = n0 + lane16;
    const float biasF = bf[col], biasI = bi[col], biasG = bc[col], biasO = bo[col];
#pragma unroll
    for (int t = 0; t < 2; ++t) {
        const int mbase = (wave * 2 + t) * 16 + half * 8;
#pragma unroll
        for (int v = 0; v < 8; ++v) {
            const int brow = mbase + v;
            const size_t idx = (size_t)brow * C_DIM + col;
            const float fv = sigmoidf_(acc[t][0][v] + biasF);
            const float iv = sigmoidf_(acc[t][1][v] + biasI);
            const float gv = tanhf(acc[t][2][v] + biasG);
            const float ov = sigmoidf_(acc[t][3][v] + biasO);
            const float cn = fv * cell[idx] + iv * gv;
            const float hn = ov * tanhf(cn);
            cell_new[idx]   = cn;
            hidden_new[idx] = hn;
        }
    }
}

// ---------------------------------------------------------------------------
// Kernel 3: logits = hidden_new @ Wout.T + bout   (written into d_out)
// 64 output columns per workgroup (4 tiles of 16) to cut activation re-reads
// from L2 4x (500 WGs x 4MB instead of 2000 x 4MB). Same double-buffered
// async pipeline and inner-loop shape as kernel 2.
// ---------------------------------------------------------------------------
__global__ __launch_bounds__(256) void out_gemm(
    const float* __restrict__ hidden_new,
    const float* __restrict__ Wout, const float* __restrict__ bout,
    float* __restrict__ logits)
{
    __shared__ float A_lds[2][256 * A_STR];
    __shared__ float B_lds[2][64 * B_STR];    // 64 cols x KSTEP, x2 buffers

    const int tid    = threadIdx.x;
    const int wave   = tid >> 5;
    const int lane   = tid & 31;
    const int half   = lane >> 4;
    const int lane16 = lane & 15;
    const int n0     = blockIdx.x * 64;

    v8f acc[2][4];
    const v8f vzero = {0.f, 0.f, 0.f, 0.f, 0.f, 0.f, 0.f, 0.f};
#pragma unroll
    for (int t = 0; t < 2; ++t)
#pragma unroll
        for (int g = 0; g < 4; ++g) acc[t][g] = vzero;

    // B staging: 64 contiguous weight rows, 4 threads/row
    const int rB = tid >> 2;      // 0..63
    const int qB = tid & 3;
    const float* wrow = Wout + (size_t)(n0 + rB) * C_DIM;
    const float* aSrc = hidden_new + (size_t)tid * C_DIM;

    auto fill = [&](int buf, int k0) {
        const float* a = aSrc + k0;
        float* ad = &A_lds[buf][tid * A_STR];
#pragma unroll
        for (int i = 0; i < KSTEP / 4; ++i) copy16_async(ad + 4 * i, a + 4 * i);
        const float* b = wrow + k0 + qB * 8;
        float* bd = &B_lds[buf][rB * B_STR + qB * 8];
        copy16_async(bd, b);
        copy16_async(bd + 4, b + 4);
    };

    fill(0, 0);
    for (int i = 0; i < NCHUNK; ++i) {
        const int buf = i & 1;
        wait_async_all();
        __syncthreads();
        if (i + 1 < NCHUNK) fill(1 - buf, (i + 1) * KSTEP);

        const float* Ab = &A_lds[buf][0];
        const float* Bb = &B_lds[buf][0];
#pragma unroll
        for (int kk = 0; kk < KSTEP / 4; ++kk) {
            const int ka = kk * 4 + 2 * half;
            v2f a0, a1, bf4[4];
            const int ar0 = (wave * 32 + lane16) * A_STR + ka;
            const int ar1 = (wave * 32 + 16 + lane16) * A_STR + ka;
            a0.x = Ab[ar0];     a0.y = Ab[ar0 + 1];
            a1.x = Ab[ar1];     a1.y = Ab[ar1 + 1];
#pragma unroll
            for (int g = 0; g < 4; ++g) {
                const int br = (g * 16 + lane16) * B_STR + ka;
                bf4[g].x = Bb[br];
                bf4[g].y = Bb[br + 1];
            }
#pragma unroll
            for (int g = 0; g < 4; ++g) {
                acc[0][g] = wmma_f32(a0, bf4[g], acc[0][g]);
                acc[1][g] = wmma_f32(a1, bf4[g], acc[1][g]);
            }
        }
    }

#pragma unroll
    for (int g = 0; g < 4; ++g) {
        const int col = n0 + g * 16 + lane16;
        const float bias = bout[col];
#pragma unroll
        for (int t = 0; t < 2; ++t) {
            const int mbase = (wave * 2 + t) * 16 + half * 8;
#pragma unroll
            for (int v = 0; v < 8; ++v) {
                const int brow = mbase + v;
                logits[(size_t)brow * VOCAB + col] = acc[t][g][v] + bias;
            }
        }
    }
}

// ---------------------------------------------------------------------------
// Kernel 4: in-place row-wise log_softmax over d_out[:, 0:VOCAB]
// ---------------------------------------------------------------------------
__global__ __launch_bounds__(256) void logsoftmax_rows(float* __restrict__ out)
{
    __shared__ float red[256];
    const int tid = threadIdx.x;
    float* p = out + (size_t)blockIdx.x * VOCAB;

    float m = -3.402823e38f;
    for (int i = tid; i < VOCAB; i += 256) m = fmaxf(m, p[i]);
    red[tid] = m;
    __syncthreads();
    for (int s = 128; s > 0; s >>= 1) {
        if (tid < s) red[tid] = fmaxf(red[tid], red[tid + s]);
        __syncthreads();
    }
    const float mx = red[0];
    __syncthreads();

    float sum = 0.f;
    for (int i = tid; i < VOCAB; i += 256) sum += __expf(p[i] - mx);
    red[tid] = sum;
    __syncthreads();
    for (int s = 128; s > 0; s >>= 1) {
        if (tid < s) red[tid] += red[tid + s];
        __syncthreads();
    }
    const float lse = mx + logf(red[0]);
    __syncthreads();

    for (int i = tid; i < VOCAB; i += 256) p[i] -= lse;
}

// ---------------------------------------------------------------------------
extern "C" void kernel_launch(void* const* d_in, const int* in_sizes, int n_in,
                              void* d_out, int out_size, void* d_ws, size_t ws_size,
                              hipStream_t stream) {
    const float* category = (const float*)d_in[0];
    const float* x        = (const float*)d_in[1];
    const float* hidden   = (const float*)d_in[2];
    const float* cell     = (const float*)d_in[3];
    const float* Wf       = (const float*)d_in[4];
    const float* bf       = (const float*)d_in[5];
    const float* Wi       = (const float*)d_in[6];
    const float* bi       = (const float*)d_in[7];
    const float* Wc       = (const float*)d_in[8];
    const float* bc       = (const float*)d_in[9];
    const float* Wo       = (const float*)d_in[10];
    const float* bo       = (const float*)d_in[11];
    const float* Wout     = (const float*)d_in[12];
    const float* bout     = (const float*)d_in[13];

    float* out = (float*)d_out;
    float* out_logits = out;                                          // [B, VOCAB]
    float* out_cell   = out + (size_t)B_DIM * VOCAB;                  // [B, C]
    float* out_hidden = out + (size_t)B_DIM * VOCAB + (size_t)B_DIM * C_DIM;

    float* xc = (float*)d_ws;                                         // [B, C] = 4 MB

    // 1) concat inputs
    concat_inputs<<<(B_DIM * C_DIM / 4) / 256, 256, 0, stream>>>(
        category, x, hidden, xc);

    // 2) fused 4-gate GEMM + LSTM update (C/16 = 256 column slices)
    lstm_gate_gemm<<<C_DIM / 16, 256, 0, stream>>>(
        xc, Wf, bf, Wi, bi, Wc, bc, Wo, bo, cell, out_cell, out_hidden);

    // 3) output projection (VOCAB/64 = 500 column slices of 64)
    out_gemm<<<VOCAB / 64, 256, 0, stream>>>(
        out_hidden, Wout, bout, out_logits);

    // 4) in-place log_softmax, one workgroup per batch row
    logsoftmax_rows<<<B_DIM, 256, 0, stream>>>(out_logits);
}